// HungarianMatcher_8650064134686
// MI455X (gfx1250) — compile-verified
//
#include <hip/hip_runtime.h>
#include <math.h>
#include <stdint.h>

// Problem constants (from reference)
#define BSZ   64
#define NQ    300
#define TGT   1280            // T = 64*20
#define NROW  (BSZ * NQ)      // 19200 rows of the cost matrix
#define ROWS  8               // rows per block
#define TPB   320             // 10 wave32s; 320*4 = 1280 targets
#define NBLK  (NROW / ROWS)   // 2400 blocks

typedef __attribute__((ext_vector_type(4))) float vfloat4;
typedef int v4i __attribute__((ext_vector_type(4)));
typedef __attribute__((address_space(1))) v4i gas_v4i;   // global (AS1) int4
typedef __attribute__((address_space(3))) v4i las_v4i;   // LDS (AS3) int4

#if __has_builtin(__builtin_amdgcn_global_load_async_to_lds_b128) && \
    __has_builtin(__builtin_amdgcn_s_wait_asynccnt)
#define USE_ASYNC_LDS 1
#endif

__global__ __launch_bounds__(TPB) void matcher_cost_kernel(
    const float* __restrict__ pred_logits,  // [NROW, 2]
    const float* __restrict__ pred_spans,   // [NROW, 2]
    const float* __restrict__ tgt_spans,    // [TGT, 2]
    const float* __restrict__ ref_points,   // [NROW, 2]
    float* __restrict__ out)                // [NROW, TGT]
{
    __shared__ float s_tgt[TGT * 2];        // 10240 B: (cx, w) per target
    __shared__ float s_row[ROWS][5];        // cx, w, x1, x2, rowconst

    const int t    = threadIdx.x;
    const int row0 = blockIdx.x * ROWS;

    // ---- Stage tgt_spans (10 KB) into LDS: 32 B per thread, async DMA path ----
#ifdef USE_ASYNC_LDS
    {
        // thread t covers bytes [32t, 32t+32)
        const char* gp = (const char*)tgt_spans + (size_t)t * 32;
        char*       lp = (char*)s_tgt + t * 32;
        // generic->AS1: full 64-bit address; generic->AS3: low 32 bits = LDS byte offset
        gas_v4i* g1 = (gas_v4i*)(uintptr_t)gp;
        las_v4i* l3 = (las_v4i*)(uint32_t)(uintptr_t)lp;
        __builtin_amdgcn_global_load_async_to_lds_b128(g1, l3, 0, 0);
        __builtin_amdgcn_global_load_async_to_lds_b128(g1, l3, 16, 0);
    }
#else
    {
        const vfloat4* g4 = (const vfloat4*)tgt_spans;   // 640 vec4s
        vfloat4* s4 = (vfloat4*)s_tgt;
        s4[2 * t]     = g4[2 * t];
        s4[2 * t + 1] = g4[2 * t + 1];
    }
#endif

    // ---- Per-row constants: first ROWS lanes compute them once ----
    if (t < ROWS) {
        const int i = row0 + t;
        const float l0 = pred_logits[2 * i];
        const float l1 = pred_logits[2 * i + 1];
        // softmax prob of class 0 = sigmoid(l0 - l1)  -> cost_class = -prob0
        const float prob0 =
            __builtin_amdgcn_rcpf(1.0f + __builtin_amdgcn_exp2f((l1 - l0) * 1.44269504088896f));
        const float cx = pred_spans[2 * i];
        const float w  = pred_spans[2 * i + 1];
        const float x1 = cx - 0.5f * w;
        const float x2 = cx + 0.5f * w;
        const float dx = x1 - ref_points[2 * i];
        const float dy = x2 - ref_points[2 * i + 1];
        const float cref = __builtin_amdgcn_sqrtf(dx * dx + dy * dy);
        s_row[t][0] = cx;
        s_row[t][1] = w;
        s_row[t][2] = x1;
        s_row[t][3] = x2;
        s_row[t][4] = cref - prob0;   // cost_reference + cost_class
    }

#ifdef USE_ASYNC_LDS
    __builtin_amdgcn_s_wait_asynccnt(0);   // my async loads have landed in LDS
#endif
    __syncthreads();                        // everyone's staging + row consts visible

    // ---- Each thread owns 4 consecutive targets, kept in VGPRs ----
    const int j0 = 4 * t;
    float tc[4], tw[4], t1[4], t2[4];
#pragma unroll
    for (int k = 0; k < 4; ++k) {
        const float cx = s_tgt[2 * (j0 + k)];
        const float w  = s_tgt[2 * (j0 + k) + 1];
        tc[k] = cx;
        tw[k] = w;
        t1[k] = cx - 0.5f * w;
        t2[k] = cx + 0.5f * w;
    }

    // ---- Sweep the block's rows, reusing target registers ----
#pragma unroll
    for (int r = 0; r < ROWS; ++r) {
        const float cx = s_row[r][0];
        const float w  = s_row[r][1];
        const float x1 = s_row[r][2];
        const float x2 = s_row[r][3];
        const float rc = s_row[r][4];

        vfloat4 res;
#pragma unroll
        for (int k = 0; k < 4; ++k) {
            // L1 span cost in (cx, w) space
            const float span = fabsf(cx - tc[k]) + fabsf(w - tw[k]);
            // 1-D GIoU in (x1, x2) space
            const float lt    = fmaxf(x1, t1[k]);
            const float rb    = fminf(x2, t2[k]);
            const float inter = fmaxf(rb - lt, 0.0f);
            const float uni   = w + tw[k] - inter;           // area_a + area_b - inter
            const float left  = fminf(x1, t1[k]);
            const float right = fmaxf(x2, t2[k]);
            const float enc   = fmaxf(right - left, 0.0f);
            const float iou   = inter * __builtin_amdgcn_rcpf(uni);
            const float giou  = iou - (enc - uni) * __builtin_amdgcn_rcpf(enc);
            res[k] = rc + span - giou;                       // + cost_class + cost_reference
        }
        const size_t row = (size_t)(row0 + r);
        __builtin_nontemporal_store(res, (vfloat4*)(out + row * (size_t)TGT + j0));
    }
}

extern "C" void kernel_launch(void* const* d_in, const int* in_sizes, int n_in,
                              void* d_out, int out_size, void* d_ws, size_t ws_size,
                              hipStream_t stream) {
    const float* pred_logits = (const float*)d_in[0];
    const float* pred_spans  = (const float*)d_in[1];
    const float* tgt_spans   = (const float*)d_in[2];
    const float* ref_points  = (const float*)d_in[3];
    float* out = (float*)d_out;

    hipLaunchKernelGGL(matcher_cost_kernel, dim3(NBLK), dim3(TPB), 0, stream,
                       pred_logits, pred_spans, tgt_spans, ref_points, out);
}